// ManualLayer_36438502539858
// MI455X (gfx1250) — compile-verified
//
#include <hip/hip_runtime.h>
#include <hip/hip_bf16.h>
#include <math.h>

typedef float v2f __attribute__((ext_vector_type(2)));
typedef float v8f __attribute__((ext_vector_type(8)));

#define BB   64          // batch
#define TT   2048        // timesteps
#define DD   1024        // input dim
#define UU   1024        // hidden dim
#define NWG  64          // workgroups (one per 16 output columns)
#define NTHR 128         // 4 waves of 32

// Native CDNA5 V_TANH_F32 if the builtin exists on this toolchain.
#if __has_builtin(__builtin_amdgcn_tanhf)
  #define TANH_F32(x) __builtin_amdgcn_tanhf(x)
#elif __has_builtin(__builtin_amdgcn_tanh_f32)
  #define TANH_F32(x) __builtin_amdgcn_tanh_f32(x)
#else
  #define TANH_F32(x) tanhf(x)
#endif

// Persistent fused Elman-RNN kernel.
// WG g owns output columns [16g, 16g+16); wave w owns batch rows [16w, 16w+16).
// Per step: 16x16 tile = tanh( x_t[16x1024] @ Wx_cols + h[16x1024] @ Wh_cols + b ),
// K accumulated with V_WMMA_F32_16X16X4_F32 (f32 in / f32 acc, matches reference math).
__global__ __launch_bounds__(NTHR, 1) void elman_rnn_persistent(
    const float* __restrict__ x,      // [B, T, D]
    const float* __restrict__ Wx,     // [D, U]
    const float* __restrict__ Wh,     // [U, U]
    const float* __restrict__ bias,   // [U]
    float*       __restrict__ out,    // [B, U] final h
    float*       __restrict__ h0,     // ws ping buffer [B, U] (zeroed by host memset)
    float*       __restrict__ h1,     // ws pong buffer [B, U]
    unsigned*    __restrict__ counter)// ws barrier counter (zeroed by host memset)
{
    // 320KB-LDS CDNA5: keep this WG's W column slices resident for all 2048 steps.
    // Layout: column-major-in-N, K-contiguous => B-fragment loads are ds_load_2addr_b64.
    __shared__ float ldsW[2 * 16 * 1024];   // 128 KB: [0,16K)=Wx cols, [16K,32K)=Wh cols

    const int tid  = threadIdx.x;
    const int lane = tid & 31;
    const int wv   = tid >> 5;
    const int g    = blockIdx.x;
    const int n0   = g * 16;          // output column base
    const int m0   = wv * 16;         // batch row base
    const int am   = lane & 15;       // M (A frag) == N (B/C frag) lane index
    const int kh   = (lane >> 4) * 2; // K sub-offset per lane half

    // ---- one-time preload of W slices into LDS (consecutive k per thread) ----
    for (int i = tid; i < 16 * 1024; i += NTHR) {
        const int n = i >> 10;
        const int k = i & 1023;
        ldsW[n * 1024 + k]             = Wx[(size_t)k * UU + (n0 + n)];
        ldsW[16 * 1024 + n * 1024 + k] = Wh[(size_t)k * UU + (n0 + n)];
    }
    __syncthreads();

    const float* pBx = &ldsW[am * 1024 + kh];             // B frag base (Wx half)
    const float* pBh = &ldsW[16 * 1024 + am * 1024 + kh]; // B frag base (Wh half)
    const float* pArow = x + (size_t)(m0 + am) * TT * DD + kh; // this lane's x row
    const float  bv = bias[n0 + am];

    for (int t = 0; t < TT; ++t) {
        const float* hin  = (t & 1) ? h1 : h0;
        float*       hout = (t == TT - 1) ? out : ((t & 1) ? h0 : h1);

        const float* pA = pArow + (size_t)t * DD;                 // x_t strip
        const float* pH = hin + (size_t)(m0 + am) * UU + kh;      // h strip

        v8f acc = {};

        // ---- x_t @ Wx (K = 0..1023, step 4) ----
        #pragma unroll 8
        for (int k0 = 0; k0 < DD; k0 += 4) {
            v2f a = *(const v2f*)(pA + k0);
            v2f b = *(const v2f*)(pBx + k0);
            acc = __builtin_amdgcn_wmma_f32_16x16x4_f32(
                false, a, false, b, (short)0, acc, false, false);
        }
        // ---- h_{t-1} @ Wh (K = 0..1023, step 4) ----
        #pragma unroll 8
        for (int k0 = 0; k0 < UU; k0 += 4) {
            v2f a = *(const v2f*)(pH + k0);
            v2f b = *(const v2f*)(pBh + k0);
            acc = __builtin_amdgcn_wmma_f32_16x16x4_f32(
                false, a, false, b, (short)0, acc, false, false);
        }

        // ---- bias + tanh + store (C layout: VGPR j -> M=j (lanes<16) / j+8) ----
        {
            float* hrow = hout + (size_t)(m0 + (lane >> 4) * 8) * UU + n0 + am;
            #pragma unroll
            for (int j = 0; j < 8; ++j) {
                hrow[(size_t)j * UU] = TANH_F32(acc[j] + bv);
            }
        }

        if (t + 1 < TT) {
            // Prefetch next timestep's x strip (global_prefetch_b8) while we sync.
            const float* pn = pArow + (size_t)(t + 1) * DD;
            #pragma unroll
            for (int off = 0; off < DD; off += 128) {
                __builtin_prefetch(pn + off, 0, 1);
            }

            // ---- device-wide step barrier (monotonic counter, agent scope) ----
            __syncthreads();
            if (tid == 0) {
                __hip_atomic_fetch_add(counter, 1u, __ATOMIC_RELEASE,
                                       __HIP_MEMORY_SCOPE_AGENT);
                const unsigned target = (unsigned)NWG * (unsigned)(t + 1);
                while (__hip_atomic_load(counter, __ATOMIC_ACQUIRE,
                                         __HIP_MEMORY_SCOPE_AGENT) < target) {
                    __builtin_amdgcn_s_sleep(1);
                }
            }
            __syncthreads();
            // all threads acquire: invalidate near caches so the new h is visible
            __builtin_amdgcn_fence(__ATOMIC_ACQUIRE, "agent");
        }
    }
}

extern "C" void kernel_launch(void* const* d_in, const int* in_sizes, int n_in,
                              void* d_out, int out_size, void* d_ws, size_t ws_size,
                              hipStream_t stream) {
    const float* x    = (const float*)d_in[0]; // [B,T,D]
    const float* Wx   = (const float*)d_in[1]; // [D,U]
    const float* Wh   = (const float*)d_in[2]; // [U,U]
    const float* bias = (const float*)d_in[3]; // [U]
    float* out = (float*)d_out;

    unsigned char* ws = (unsigned char*)d_ws;
    unsigned* counter = (unsigned*)ws;                       // 4 B (256 B reserved)
    float* h0 = (float*)(ws + 256);                          // 256 KB
    float* h1 = (float*)(ws + 256 + (size_t)BB * UU * 4);    // 256 KB

    // zero barrier counter + h0 (h_{-1} = 0) each launch; graph-capture legal
    (void)hipMemsetAsync(ws, 0, 256 + (size_t)BB * UU * sizeof(float), stream);

    elman_rnn_persistent<<<dim3(NWG), dim3(NTHR), 0, stream>>>(
        x, Wx, Wh, bias, out, h0, h1, counter);
}